// LocalWindowTripleLoss_33973191311851
// MI455X (gfx1250) — compile-verified
//
#include <hip/hip_runtime.h>
#include <hip/hip_bf16.h>

typedef __attribute__((ext_vector_type(2))) float v2f;
typedef __attribute__((ext_vector_type(8))) float v8f;

#define HW    16384
#define WD    128
#define CIN   64
#define COUT  256
#define BATCH 4
#define KW    11
#define STR   5
#define HO    24
#define NWIN  576   /* 24*24 */
#define NK    121
#define LPITCH 20   /* LDS pitch (floats) for transposed B panel: bank-conflict-free */

/* workspace layout in floats */
#define X1_OFF    0
#define X2_OFF    ((size_t)BATCH * COUT * HW)           /* 16,777,216 */
#define SCALE_OFF (2 * (size_t)BATCH * COUT * HW)       /* 33,554,432 */
#define SHIFT_OFF (SCALE_OFF + COUT)
#define TRIP_OFF  (SHIFT_OFF + COUT)
#define VALID_OFF (TRIP_OFF + BATCH * NK)

// ---------------------------------------------------------------------------
// Zero atomic accumulators (deterministic per launch; graph-replay safe)
// ---------------------------------------------------------------------------
__global__ void k_init(float* __restrict__ trip, float* __restrict__ valid) {
  int t = blockIdx.x * blockDim.x + threadIdx.x;
  if (t < BATCH * NK) trip[t] = 0.f;
  if (t < BATCH)      valid[t] = 0.f;
}

// ---------------------------------------------------------------------------
// Fused 1x1-conv GEMM with fp32 WMMA (V_WMMA_F32_16X16X4_F32).
//   D[m,n] = sum_k W[m,k] * f(B[k,n]) + bias[m],  f = relu(bn(.)) if BNRELU
// Workgroup = 8 waves: 4 along M (64 rows), 2 along N (2 x 64 cols).
// The 16k x 128n B panel is staged *transposed* into LDS via the per-lane
// async scatter path (global_load_async_to_lds_b32, ASYNCcnt), double
// buffered. Column-major LDS makes each WMMA B fragment (per-lane K pair)
// one contiguous ds_load_b64 -> no register marshaling in the hot loop.
// Global reads stay coalesced: each wave reads 32 consecutive columns of a
// row; the async unit performs the transpose on the LDS-write side.
// ---------------------------------------------------------------------------
template <int KDIM, bool BNRELU, bool OUT_NHWC>
__global__ __launch_bounds__(256)
void k_gemm(const float* __restrict__ Bg,    // [B][KDIM][HW]
            const float* __restrict__ Wg,    // [COUT][KDIM] row-major
            const float* __restrict__ bias,  // [COUT]
            const float* __restrict__ scale, // [KDIM] (BNRELU only)
            const float* __restrict__ shift, // [KDIM] (BNRELU only)
            float* __restrict__ Out) {
  __shared__ float sB[2][128][LPITCH];       // 2 x 10KB, column-major panel

  const int t = threadIdx.x, lane = t & 31, wv = t >> 5;
  const int b = blockIdx.z;
  const int m_base = blockIdx.y * 64 + (wv & 3) * 16;
  const int n_blk  = blockIdx.x * 128;
  const int n_off  = (wv >> 2) * 64;
  const float* __restrict__ Bm = Bg + (size_t)b * KDIM * HW + n_blk;
  const unsigned lds0 = (unsigned)(uintptr_t)(&sB[0][0][0]); // LDS byte offset

  v8f acc[4] = {};
  const int ar  = m_base + (lane & 15);   // A row (M)
  const int kh  = (lane >> 4) << 1;       // K sub-offset 0 / 2 per half-wave
  const int col = n_off + (lane & 15);

  // ---- prologue: stage 0 -> buffer 0 (8 x b32 scatter per thread) ----
#pragma unroll
  for (int i = 0; i < 8; ++i) {
    int l = t + i * 256;                  // element id: row = l>>7 (0..15), col = l&127
    int row = l >> 7, cc = l & 127;
    const float* g = Bm + (size_t)row * HW + cc;
    unsigned lds = lds0 + (unsigned)((cc * LPITCH + row) * 4);
    asm volatile("global_load_async_to_lds_b32 %0, %1, off"
                 :: "v"(lds), "v"(g) : "memory");
  }

  const int NST = KDIM / 16;
  for (int s = 0; s < NST; ++s) {
    if (s + 1 < NST) {
      // issue next stage into the other buffer, then wait only for stage s
#pragma unroll
      for (int i = 0; i < 8; ++i) {
        int l = t + i * 256;
        int row = l >> 7, cc = l & 127;
        const float* g = Bm + (size_t)((s + 1) * 16 + row) * HW + cc;
        unsigned lds = lds0 +
            (unsigned)((((s + 1) & 1) * 128 * LPITCH + cc * LPITCH + row) * 4);
        asm volatile("global_load_async_to_lds_b32 %0, %1, off"
                     :: "v"(lds), "v"(g) : "memory");
      }
      asm volatile("s_wait_asynccnt 0x8" ::: "memory"); // in-order: stage s done
    } else {
      asm volatile("s_wait_asynccnt 0x0" ::: "memory");
    }
    __syncthreads();                                    // cross-wave visibility

    const float(*__restrict__ buf)[LPITCH] = sB[s & 1];
#pragma unroll
    for (int r = 0; r < 16; r += 4) {
      const int kk = s * 16 + r + kh;
      v2f a = *(const v2f*)(Wg + (size_t)ar * KDIM + kk); // 8B-aligned (kk even)
      float s0 = 0.f, h0 = 0.f, s1 = 0.f, h1 = 0.f;
      if (BNRELU) { s0 = scale[kk]; h0 = shift[kk]; s1 = scale[kk + 1]; h1 = shift[kk + 1]; }
#pragma unroll
      for (int j = 0; j < 4; ++j) {
        // per-lane K pair is contiguous in the transposed panel: one b64 load
        v2f bf = *(const v2f*)&buf[col + j * 16][r + kh];
        if (BNRELU) {
          bf.x = fmaxf(bf.x * s0 + h0, 0.f);
          bf.y = fmaxf(bf.y * s1 + h1, 0.f);
        }
        acc[j] = __builtin_amdgcn_wmma_f32_16x16x4_f32(
            false, a, false, bf, (short)0, acc[j], false, false);
      }
    }
    __syncthreads();  // all reads of buf done before it is re-targeted
  }

  // C/D layout: vgpr r, lane L -> M = r + 8*(L/16), N = L%16
#pragma unroll
  for (int j = 0; j < 4; ++j) {
    int n = n_blk + n_off + j * 16 + (lane & 15);
#pragma unroll
    for (int r = 0; r < 8; ++r) {
      int m = m_base + r + ((lane >> 4) << 3);
      if (OUT_NHWC)
        Out[((size_t)b * HW + n) * COUT + m] = acc[j][r] + bias[m];
      else
        Out[((size_t)b * COUT + m) * HW + n] = acc[j][r] + bias[m];
    }
  }
}

// ---------------------------------------------------------------------------
// BN batch statistics -> fused scale/shift (gamma/sqrt(var+eps), beta - mean*s)
// One block per channel.
// ---------------------------------------------------------------------------
__global__ __launch_bounds__(256)
void k_bnstats(const float* __restrict__ x1, const float* __restrict__ gamma,
               const float* __restrict__ beta, float* __restrict__ scale,
               float* __restrict__ shift) {
  const int c = blockIdx.x, t = threadIdx.x;
  float s = 0.f, q = 0.f;
  for (int b = 0; b < BATCH; ++b) {
    const float* p = x1 + ((size_t)b * COUT + c) * HW;
    for (int i = t; i < HW; i += 256) { float v = p[i]; s += v; q += v * v; }
  }
  __shared__ float rs[256], rq[256];
  rs[t] = s; rq[t] = q; __syncthreads();
  for (int o = 128; o > 0; o >>= 1) {
    if (t < o) { rs[t] += rs[t + o]; rq[t] += rq[t + o]; }
    __syncthreads();
  }
  if (t == 0) {
    const float inv_n = 1.f / (float)(BATCH * HW);
    float mean = rs[0] * inv_n;
    float var  = rq[0] * inv_n - mean * mean;
    float sc   = gamma[c] * rsqrtf(var + 1e-5f);
    scale[c] = sc;
    shift[c] = beta[c] - mean * sc;
  }
}

// ---------------------------------------------------------------------------
// Window triple loss. One 256-thread block per (b, ho, wo) window.
// x2 is NHWC so all channel accesses are coalesced.
// ---------------------------------------------------------------------------
__global__ __launch_bounds__(256)
void k_winloss(const float* __restrict__ sal, const float* __restrict__ x2,
               float* __restrict__ trip_acc, float* __restrict__ valid_acc) {
  const int t = threadIdx.x, lane = t & 31, wv = t >> 5;
  const int b = blockIdx.y;
  const int ho = blockIdx.x / HO, wo = blockIdx.x % HO;
  const int h0 = ho * STR, w0 = wo * STR;

  __shared__ float sm_m[NK];
  __shared__ float sm_pc[COUT], sm_nc[COUT];
  __shared__ float sm_cnt[2];

  // binarized mask for the 121 taps (identity resize: same spatial size)
  if (t < NK) {
    int i = t / KW, j = t % KW;
    float s = sal[((size_t)b * WD + (h0 + i)) * WD + (w0 + j)];
    sm_m[t] = (s > 0.5f) ? 1.f : 0.f;
  }
  __syncthreads();
  if (t == 0) {
    float p = 0.f;
    for (int k = 0; k < NK; ++k) p += sm_m[k];
    sm_cnt[0] = p; sm_cnt[1] = (float)NK - p;
  }
  __syncthreads();
  const float poscnt = sm_cnt[0], negcnt = sm_cnt[1];
  const bool valid = (poscnt > 0.5f) && (negcnt > 0.5f);

  const float* __restrict__ base = x2 + (size_t)b * HW * COUT;

  // per-channel masked centroids: thread t == channel c (coalesced NHWC loads)
  float ps = 0.f, ns = 0.f;
  for (int k = 0; k < NK; ++k) {
    int n = (h0 + k / KW) * WD + (w0 + k % KW);
    float f = base[(size_t)n * COUT + t];
    float mk = sm_m[k];
    ps += mk * f;
    ns += (1.f - mk) * f;
  }
  sm_pc[t] = ps / (poscnt + 1e-6f);
  sm_nc[t] = ns / (negcnt + 1e-6f);
  __syncthreads();

  if (valid) {
    for (int k = wv; k < NK; k += 8) {
      int n = (h0 + k / KW) * WD + (w0 + k % KW);
      const float* fp = base + (size_t)n * COUT;
      float dp = 0.f, dn = 0.f;
#pragma unroll
      for (int c = lane; c < COUT; c += 32) {
        float f  = fp[c];
        float d1 = sm_pc[c] - f, d2 = sm_nc[c] - f;
        dp += d1 * d1;
        dn += d2 * d2;
      }
#pragma unroll
      for (int o = 16; o > 0; o >>= 1) {   // wave32 butterfly reduction
        dp += __shfl_xor(dp, o, 32);
        dn += __shfl_xor(dn, o, 32);
      }
      if (lane == 0) {
        float dpc = 1.f - __expf(-0.01f * dp);
        float dnc = 1.f - __expf(-0.01f * dn);
        float mk  = sm_m[k];
        float pos_d = (mk > 0.5f) ? dpc : dnc;
        float neg_d = (mk > 0.5f) ? dnc : dpc;
        float tr = fmaxf(pos_d - neg_d + 0.5f, 0.f);
        atomicAdd(&trip_acc[b * NK + k], tr);
      }
    }
    if (t == 0) atomicAdd(&valid_acc[b], 1.f);
  }
}

// ---------------------------------------------------------------------------
// Final scalar: mean over (b,k) of trip[b,k] / (valid[b] + 1e-6)
// ---------------------------------------------------------------------------
__global__ __launch_bounds__(256)
void k_final(const float* __restrict__ trip, const float* __restrict__ valid,
             float* __restrict__ out) {
  const int t = threadIdx.x;
  float s = 0.f;
  for (int i = t; i < BATCH * NK; i += 256) s += trip[i] / (valid[i / NK] + 1e-6f);
  __shared__ float red[256];
  red[t] = s; __syncthreads();
  for (int o = 128; o > 0; o >>= 1) {
    if (t < o) red[t] += red[t + o];
    __syncthreads();
  }
  if (t == 0) out[0] = red[0] / (float)(BATCH * NK);
}

// ---------------------------------------------------------------------------
extern "C" void kernel_launch(void* const* d_in, const int* in_sizes, int n_in,
                              void* d_out, int out_size, void* d_ws, size_t ws_size,
                              hipStream_t stream) {
  const float* sal   = (const float*)d_in[0];
  const float* feat  = (const float*)d_in[1];
  const float* w1    = (const float*)d_in[2];
  const float* b1    = (const float*)d_in[3];
  const float* gamma = (const float*)d_in[4];
  const float* beta  = (const float*)d_in[5];
  const float* w2    = (const float*)d_in[6];
  const float* b2    = (const float*)d_in[7];

  float* ws    = (float*)d_ws;
  float* x1    = ws + X1_OFF;
  float* x2    = ws + X2_OFF;
  float* scale = ws + SCALE_OFF;
  float* shift = ws + SHIFT_OFF;
  float* trip  = ws + TRIP_OFF;
  float* valid = ws + VALID_OFF;
  float* out   = (float*)d_out;

  k_init<<<2, 256, 0, stream>>>(trip, valid);
  k_gemm<CIN, false, false>
      <<<dim3(HW / 128, COUT / 64, BATCH), 256, 0, stream>>>(feat, w1, b1, scale, shift, x1);
  k_bnstats<<<COUT, 256, 0, stream>>>(x1, gamma, beta, scale, shift);
  k_gemm<COUT, true, true>
      <<<dim3(HW / 128, COUT / 64, BATCH), 256, 0, stream>>>(x1, w2, b2, scale, shift, x2);
  k_winloss<<<dim3(NWIN, BATCH), 256, 0, stream>>>(sal, x2, trip, valid);
  k_final<<<1, 256, 0, stream>>>(trip, valid, out);
}